// FeatureAlign_55559696941290
// MI455X (gfx1250) — compile-verified
//
#include <hip/hip_runtime.h>

typedef __attribute__((ext_vector_type(2))) float v2f;
typedef __attribute__((ext_vector_type(8))) float v8f;

#define Hc 64
#define Wc 64
#define Dc 64
#define HWc 4096

// ---------------------------------------------------------------------------
// Kernel 0: zero the nonzero-counters in workspace (16 uints)
// ---------------------------------------------------------------------------
__global__ __launch_bounds__(32) void fa_init(unsigned* __restrict__ cnts) {
  if (threadIdx.x < 16) cnts[threadIdx.x] = 0u;
}

// ---------------------------------------------------------------------------
// Kernel 1: per-batch nonzero counts of cur_feat / prev_feat
// ---------------------------------------------------------------------------
__global__ __launch_bounds__(256) void fa_counts(const float* __restrict__ cur,
                                                 const float* __restrict__ prev,
                                                 unsigned* __restrict__ cnts) {
  const int n = blockIdx.y;
  const size_t base = (size_t)n * Dc * HWc;
  unsigned c0 = 0, c1 = 0;
  const int stride = gridDim.x * blockDim.x;
  for (int e = blockIdx.x * blockDim.x + threadIdx.x; e < Dc * HWc; e += stride) {
    c0 += (cur[base + e]  != 0.0f) ? 1u : 0u;
    c1 += (prev[base + e] != 0.0f) ? 1u : 0u;
  }
  // wave32 reduction
  #pragma unroll
  for (int off = 16; off > 0; off >>= 1) {
    c0 += __shfl_down(c0, off, 32);
    c1 += __shfl_down(c1, off, 32);
  }
  if ((threadIdx.x & 31) == 0) {
    atomicAdd(&cnts[n], c0);
    atomicAdd(&cnts[8 + n], c1);
  }
}

// ---------------------------------------------------------------------------
// Kernel 2: fused windowed-affinity + normalize + gather-apply.
// One wave per (n, y, 16-wide x tile).
// ---------------------------------------------------------------------------
__global__ __launch_bounds__(32) void fa_main(const float* __restrict__ cur,
                                              const float* __restrict__ prev,
                                              const float* __restrict__ pm,
                                              float* __restrict__ out,
                                              const unsigned* __restrict__ cnts) {
  const int lane = threadIdx.x;  // 0..31
  const int i    = lane & 15;    // column index within 16-wide tile
  const int hi   = lane >> 4;    // half-wave select (K / M-row half)
  const int x0   = blockIdx.x * 16;
  const int y    = blockIdx.y;
  const int n    = blockIdx.z;

  __shared__ float aff[5][20][16];       // [win row][q col m=0..19][p col]
  __shared__ float wlds[25][16];         // normalized weights per p
  __shared__ float band[Dc * 100];       // prev_mem band: [d][5 rows][20 cols]

  const float* curN  = cur  + (size_t)n * Dc * HWc;
  const float* prevN = prev + (size_t)n * Dc * HWc;
  const float* pmN   = pm   + (size_t)n * Dc * HWc;
  float*       outN  = out  + (size_t)n * Dc * HWc;

  // ---- cooperative stage of prev_mem window band into LDS (clamped addrs)
  for (int t = lane; t < Dc * 100; t += 32) {
    int d  = t / 100;
    int rr = (t / 20) % 5;
    int mm = t % 20;
    int yq = min(max(y + rr - 2, 0), Hc - 1);
    int xq = min(max(x0 + mm - 2, 0), Wc - 1);
    band[t] = pmN[(size_t)d * HWc + yq * Wc + xq];
  }

  // ---- preload cur tile as WMMA B operand: B[K=d][N=p], lane holds N=i,
  //      K = 4k + 2*hi + {0,1}
  v2f bcur[16];
  {
    const float* cp0 = curN + y * Wc + x0 + i;
    #pragma unroll
    for (int k = 0; k < 16; ++k) {
      int d = 4 * k + 2 * hi;
      bcur[k].x = cp0[(size_t)d * HWc];
      bcur[k].y = cp0[(size_t)(d + 1) * HWc];
    }
  }

  float scale;
  {
    float cc = (float)cnts[n]     + 1e-8f;
    float cp = (float)cnts[8 + n] + 1e-8f;
    scale = 1.0f / (cc * cp);
  }

  // ---- phase 1: affinity tiles C(M=q, N=p) = sum_d prev[d,q]*cur[d,p]
  // window rows r=0..4; q columns m=0..19 covered by two 16-col tiles
  for (int r = 0; r < 5; ++r) {
    int yc = min(max(y + r - 2, 0), Hc - 1);
    #pragma unroll
    for (int t = 0; t < 2; ++t) {
      // A operand: lane holds M=i (q row of tile), K = 4k + 2*hi + {0,1}
      int qx  = x0 - 2 + 4 * t + i;
      int qxc = min(max(qx, 0), Wc - 1);
      const float* ap0 = prevN + yc * Wc + qxc;
      v8f acc = {};
      #pragma unroll
      for (int k = 0; k < 16; ++k) {
        int d = 4 * k + 2 * hi;
        v2f a;
        a.x = ap0[(size_t)d * HWc];
        a.y = ap0[(size_t)(d + 1) * HWc];
        acc = __builtin_amdgcn_wmma_f32_16x16x4_f32(
            /*neg_a=*/false, a, /*neg_b=*/false, bcur[k],
            /*c_mod=*/(short)0, acc, /*reuse_a=*/false, /*reuse_b=*/false);
      }
      // C layout: lane holds N=i, rows M = 8*hi + g  (g = accumulator index)
      if (t == 0) {
        #pragma unroll
        for (int g = 0; g < 8; ++g) aff[r][8 * hi + g][i] = acc[g];
      } else if (hi == 1) {
        // tile2 global m = (8*hi + g) + 4; only m=16..19 are new
        #pragma unroll
        for (int g = 4; g < 8; ++g) aff[r][12 + g][i] = acc[g];
      }
    }
  }
  __syncthreads();

  // ---- phase 2: relu, scale, validity mask, mass-normalize (lanes 0..15)
  if (lane < 16) {
    int x = x0 + i;
    float wv[25];
    float mass = 0.0f;
    #pragma unroll
    for (int r = 0; r < 5; ++r) {
      int yy = y + r - 2;
      bool vy = (yy >= 0) && (yy < Hc);
      #pragma unroll
      for (int wc = 0; wc < 5; ++wc) {
        int xx = x + wc - 2;
        bool vx = (xx >= 0) && (xx < Wc);
        float a = aff[r][i + wc][i] * scale;
        a = fmaxf(a, 0.0f);
        a = (vy && vx) ? a : 0.0f;
        wv[r * 5 + wc] = a;
        mass += a;
      }
    }
    bool zm  = fabsf(mass) < 1e-7f;
    float inv = zm ? 0.0f : (1.0f / mass);
    #pragma unroll
    for (int k = 0; k < 25; ++k) {
      // zero-mass passthrough == center-delta weights (center always valid)
      float w = zm ? ((k == 12) ? 1.0f : 0.0f) : wv[k] * inv;
      wlds[k][i] = w;
    }
  }
  __syncthreads();

  // ---- phase 3: out[d,p] = sum_k w[p,k] * pm_band[d, q(p,k)]
  #pragma unroll
  for (int dc = 0; dc < 4; ++dc) {
    float acc[8] = {0.f,0.f,0.f,0.f,0.f,0.f,0.f,0.f};
    for (int k = 0; k < 25; ++k) {
      int r = k / 5, wc = k % 5;
      float wk = wlds[k][i];
      const float* bp = &band[r * 20 + (i + wc)];
      #pragma unroll
      for (int dd = 0; dd < 8; ++dd) {
        int d = hi + 2 * (dc * 8 + dd);
        acc[dd] += wk * bp[d * 100];
      }
    }
    #pragma unroll
    for (int dd = 0; dd < 8; ++dd) {
      int d = hi + 2 * (dc * 8 + dd);
      outN[(size_t)d * HWc + y * Wc + x0 + i] = acc[dd];
    }
  }
}

// ---------------------------------------------------------------------------
extern "C" void kernel_launch(void* const* d_in, const int* in_sizes, int n_in,
                              void* d_out, int out_size, void* d_ws, size_t ws_size,
                              hipStream_t stream) {
  const float* cur  = (const float*)d_in[0];
  const float* prev = (const float*)d_in[1];
  const float* pmem = (const float*)d_in[2];
  float* out = (float*)d_out;
  unsigned* cnts = (unsigned*)d_ws;

  fa_init<<<1, 32, 0, stream>>>(cnts);
  fa_counts<<<dim3(32, 8), 256, 0, stream>>>(cur, prev, cnts);
  fa_main<<<dim3(Wc / 16, Hc, 8), 32, 0, stream>>>(cur, prev, pmem, out, cnts);
}